// GSOT_7825430413467
// MI455X (gfx1250) — compile-verified
//
#include <hip/hip_runtime.h>

// Problem constants from the reference: S=4, R=16, NT=128 -> 8192 elements, scalar f32 out.
#define ETA_F 0.003f
constexpr int kS = 4;
constexpr int kR = 16;
constexpr int kNT = 128;
constexpr int kN = kS * kR * kNT;        // 8192
constexpr int kThreads = 256;             // 8 wave32s on one WGP
constexpr int kPerThread = kN / kThreads; // 32 consecutive elements per thread (stays in one row)

typedef __attribute__((ext_vector_type(2))) float v2f;
typedef __attribute__((ext_vector_type(8))) float v8f;

// Exact FP32 32-lane reduction via V_WMMA_F32_16X16X4_F32.
// A (16x4 f32): lane L<16 -> A[L][0]=vgpr0, A[L][1]=vgpr1; lane L>=16 -> K=2,3.
// We set vgpr0 = x, vgpr1 = 0, B = all ones, C = 0:
//   D[m][n] = x_m * 1 + x_{m+16} * 1   (rows of A at K=1,3 are zero)
// C/D layout: lane L, vgpr r holds D[r + 8*(L/16)][L%16].
// So sum of a lane's 8 D-vgprs = sum over half the rows; adding the value from
// lane L^16 yields the full 32-lane total on every lane. Multiplies by 1.0 and
// FP32 adds are exact, so this matches a plain FP32 tree sum.
__device__ inline float wave_reduce_sum_wmma(float x) {
  v2f a;
  a[0] = x;
  a[1] = 0.0f;
  v2f b;
  b[0] = 1.0f;
  b[1] = 1.0f;
  v8f c = {};
  c = __builtin_amdgcn_wmma_f32_16x16x4_f32(
      /*neg_a=*/false, a, /*neg_b=*/false, b,
      /*c_mod=*/(short)0, c, /*reuse_a=*/false, /*reuse_b=*/false);
  float s = ((c[0] + c[1]) + (c[2] + c[3])) + ((c[4] + c[5]) + (c[6] + c[7]));
  s += __shfl_xor(s, 16, 32); // combine the two row-halves (wave32)
  return s;
}

__global__ __launch_bounds__(kThreads) void GSOT_kernel(
    const float* __restrict__ y_pred,
    const float* __restrict__ y_true,
    const int* __restrict__ col_ind,
    float* __restrict__ out) {
  __shared__ float wave_sums[kThreads / 32];

  const int t = threadIdx.x;
  const int base = t * kPerThread;   // 32 consecutive elements; 32 | 128 so one row only
  const int row = base / kNT;        // (s,r) row index
  const int tpos0 = base % kNT;      // position-in-row of this thread's first element
  const float* __restrict__ yt_row = y_true + row * kNT;

  float acc = 0.0f;
#pragma unroll
  for (int j = 0; j < kPerThread; j += 4) {
    const float4 p = *reinterpret_cast<const float4*>(y_pred + base + j);   // global_load_b128
    const int4 ci = *reinterpret_cast<const int4*>(col_ind + base + j);     // global_load_b128
    {
      const float d = p.x - yt_row[ci.x];
      const float sh = (float)(tpos0 + j + 0 - ci.x);
      acc += ETA_F * sh * sh + d * d;
    }
    {
      const float d = p.y - yt_row[ci.y];
      const float sh = (float)(tpos0 + j + 1 - ci.y);
      acc += ETA_F * sh * sh + d * d;
    }
    {
      const float d = p.z - yt_row[ci.z];
      const float sh = (float)(tpos0 + j + 2 - ci.z);
      acc += ETA_F * sh * sh + d * d;
    }
    {
      const float d = p.w - yt_row[ci.w];
      const float sh = (float)(tpos0 + j + 3 - ci.w);
      acc += ETA_F * sh * sh + d * d;
    }
  }

  // Every wave reduces its 32 lanes with the FP32 WMMA (EXEC all-ones: full
  // block, no divergence up to here). Deterministic order -> identical output
  // on every replay.
  const float wsum = wave_reduce_sum_wmma(acc);
  if ((t & 31) == 0) wave_sums[t >> 5] = wsum;
  __syncthreads();

  if (t == 0) {
    float total = 0.0f;
#pragma unroll
    for (int w = 0; w < kThreads / 32; ++w) total += wave_sums[w];
    out[0] = total;
  }
}

extern "C" void kernel_launch(void* const* d_in, const int* in_sizes, int n_in,
                              void* d_out, int out_size, void* d_ws, size_t ws_size,
                              hipStream_t stream) {
  (void)in_sizes; (void)n_in; (void)out_size; (void)d_ws; (void)ws_size;
  const float* y_pred = (const float*)d_in[0];
  const float* y_true = (const float*)d_in[1];
  const int* col_ind = (const int*)d_in[2];  // jnp downcasts the int64 host array to int32
  float* out = (float*)d_out;
  // Tiny, latency-bound problem: one WGP-resident workgroup does everything.
  GSOT_kernel<<<1, kThreads, 0, stream>>>(y_pred, y_true, col_ind, out);
}